// RSEncoderLayer_26654567039543
// MI455X (gfx1250) — compile-verified
//
#include <hip/hip_runtime.h>
#include <hip/hip_bf16.h>

// Problem constants (match reference)
#define NN   100000
#define EE   1600000
#define IND  128
#define OUTD 64
#define TT   8

typedef float v2f __attribute__((ext_vector_type(2)));
typedef float v8f __attribute__((ext_vector_type(8)));

// ---------------------------------------------------------------------------
// 1) deg[i] = 1.0 (self loop)
__global__ __launch_bounds__(256) void k_init_deg(float* __restrict__ deg) {
    int i = blockIdx.x * 256 + threadIdx.x;
    if (i < NN) deg[i] = 1.0f;
}

// 2) deg[dst] += 1 per edge
__global__ __launch_bounds__(256) void k_accum_deg(const int* __restrict__ ei,
                                                   float* __restrict__ deg) {
    int e = blockIdx.x * 256 + threadIdx.x;
    if (e < EE) atomicAdd(&deg[ei[EE + e]], 1.0f);
}

// 3) deg -> 1/sqrt(deg) in place
__global__ __launch_bounds__(256) void k_deg_to_dinv(float* __restrict__ deg) {
    int i = blockIdx.x * 256 + threadIdx.x;
    if (i < NN) deg[i] = rsqrtf(fmaxf(deg[i], 1.0f));
}

// ---------------------------------------------------------------------------
// 4) h = x @ W using V_WMMA_F32_16X16X4_F32.
//    Block = 128 threads = 4 waves. Block covers a 16-row M tile; wave w covers
//    N-tile [16w, 16w+16). W (128x64, 32KB) + x tile (16x128, 8KB) staged in LDS.
//
//    f32 A 16x4 layout (ISA 7.12.2): lanes 0-15 hold K=0(V0),K=1(V1);
//    lanes 16-31 hold K=2(V0),K=3(V1), row M = lane%16.
//    f32 C/D 16x16: VGPR r, lanes 0-15 -> M=r; lanes 16-31 -> M=r+8; N = lane%16.
__global__ __launch_bounds__(128) void k_gemm_wmma(const float* __restrict__ x,
                                                   const float* __restrict__ W,
                                                   float* __restrict__ h) {
    __shared__ float xs[16 * IND];    // 8 KB
    __shared__ float ws[IND * OUTD];  // 32 KB
    const int t  = threadIdx.x;
    const int m0 = blockIdx.x * 16;

    #pragma unroll
    for (int i = 0; i < 16; ++i) {            // 2048 floats, coalesced
        int idx = t + i * 128;
        xs[idx] = x[m0 * IND + idx];
    }
    #pragma unroll
    for (int i = 0; i < 64; ++i) {            // 8192 floats, coalesced
        int idx = t + i * 128;
        ws[idx] = W[idx];
    }
    __syncthreads();

    const int lane  = t & 31;
    const int wave  = t >> 5;       // 0..3 -> N tile
    const int n0    = wave * 16;
    const int row   = lane & 15;    // M for A frag, N for B/C frags
    const int khalf = lane >> 4;    // selects K pair {0,1} vs {2,3}

    v8f c = {};
    #pragma unroll
    for (int kk = 0; kk < IND; kk += 4) {
        v2f a, b;
        a.x = xs[row * IND + kk + 2 * khalf + 0];
        a.y = xs[row * IND + kk + 2 * khalf + 1];
        b.x = ws[(kk + 2 * khalf + 0) * OUTD + n0 + row];
        b.y = ws[(kk + 2 * khalf + 1) * OUTD + n0 + row];
        c = __builtin_amdgcn_wmma_f32_16x16x4_f32(
                /*neg_a=*/false, a, /*neg_b=*/false, b,
                /*c_mod=*/(short)0, c, /*reuse_a=*/false, /*reuse_b=*/false);
    }

    #pragma unroll
    for (int r = 0; r < 8; ++r) {
        int m = r + 8 * khalf;
        h[(m0 + m) * OUTD + n0 + row] = c[r];
    }
}

// ---------------------------------------------------------------------------
// 5) g[i,c] = dinv[i]^2 * h[i,c] + b[c]   (self-loop term + bias)
__global__ __launch_bounds__(256) void k_init_g(const float* __restrict__ dinv,
                                                const float* __restrict__ h,
                                                const float* __restrict__ b,
                                                float* __restrict__ g) {
    long idx = (long)blockIdx.x * 256 + threadIdx.x;
    if (idx < (long)NN * OUTD) {
        int i = (int)(idx >> 6);
        int c = (int)(idx & 63);
        float di = dinv[i];
        g[idx] = di * di * h[idx] + b[c];
    }
}

// 6) per-edge scatter: g[dst,c] += dinv[src]*dinv[dst] * h[src,c]
//    64 consecutive threads handle one edge's 64 channels -> coalesced gather
//    of h[src] and coalesced f32 atomic adds into g[dst].
__global__ __launch_bounds__(256) void k_scatter(const int* __restrict__ ei,
                                                 const float* __restrict__ dinv,
                                                 const float* __restrict__ h,
                                                 float* __restrict__ g) {
    long gid = (long)blockIdx.x * 256 + threadIdx.x;
    long e   = gid >> 6;
    int  c   = (int)(gid & 63);
    if (e < EE) {
        int s = ei[e];
        int d = ei[EE + e];
        float w = dinv[s] * dinv[d];
        atomicAdd(&g[(long)d * OUTD + c], w * h[(long)s * OUTD + c]);
    }
}

// ---------------------------------------------------------------------------
// 7) Lorentz spiking dynamics, T=8 steps fused; one thread per node.
//    h row and z state live in VGPRs. Spike test arccosh(z0) >= 1  <=>
//    z0 >= cosh(1), so no inverse transcendental needed.
__global__ __launch_bounds__(256) void k_lorentz(const float* __restrict__ g,
                                                 float* __restrict__ o_seq,
                                                 float* __restrict__ z_seq) {
    int node = blockIdx.x * 256 + threadIdx.x;
    if (node >= NN) return;

    const float EPSf  = 1e-7f;
    const float COSH1 = 1.5430806348152437f;  // cosh(1)

    float hr[OUTD];
    float z[OUTD];
    #pragma unroll
    for (int c = 0; c < OUTD; ++c) hr[c] = g[(long)node * OUTD + c];
    #pragma unroll
    for (int c = 0; c < OUTD; ++c) z[c] = 0.0f;
    z[0] = 1.0f;

    for (int t = 0; t < TT; ++t) {
        // alpha = <z, h>_L = -z0*h0 + sum_{c>0} zc*hc
        float alpha = -z[0] * hr[0];
        #pragma unroll
        for (int c = 1; c < OUTD; ++c) alpha += z[c] * hr[c];

        // uu = <u,u>_L with u = h + alpha*z (recomputed to save VGPRs)
        float u0 = hr[0] + alpha * z[0];
        float uu = -u0 * u0;
        #pragma unroll
        for (int c = 1; c < OUTD; ++c) {
            float uc = hr[c] + alpha * z[c];
            uu += uc * uc;
        }
        float un   = sqrtf(fmaxf(uu, EPSf));
        float e    = __expf(un);
        float einv = __expf(-un);
        float ch   = 0.5f * (e + einv);
        float shun = 0.5f * (e - einv) / un;   // sinh(un)/un

        #pragma unroll
        for (int c = 0; c < OUTD; ++c) {
            float uc = hr[c] + alpha * z[c];
            z[c] = ch * z[c] + shun * uc;
        }

        float z0    = fmaxf(z[0], 1.0f + EPSf);
        bool  spike = (z0 >= COSH1);
        o_seq[(long)t * NN + node] = spike ? 1.0f : 0.0f;
        if (spike) {
            #pragma unroll
            for (int c = 0; c < OUTD; ++c) z[c] = 0.0f;
            z[0] = 1.0f;
        }
        float4* dst = (float4*)&z_seq[(long)t * NN * OUTD + (long)node * OUTD];
        #pragma unroll
        for (int q = 0; q < OUTD / 4; ++q)
            dst[q] = make_float4(z[4 * q], z[4 * q + 1], z[4 * q + 2], z[4 * q + 3]);
    }
}

// ---------------------------------------------------------------------------
extern "C" void kernel_launch(void* const* d_in, const int* in_sizes, int n_in,
                              void* d_out, int out_size, void* d_ws, size_t ws_size,
                              hipStream_t stream) {
    const float* x  = (const float*)d_in[0];   // [N,128]
    const int*   ei = (const int*)d_in[1];     // [2,E]
    const float* W  = (const float*)d_in[2];   // [128,64]
    const float* b  = (const float*)d_in[3];   // [64]

    float* wsf  = (float*)d_ws;
    float* deg  = wsf;                          // [N]  (becomes dinv in place)
    float* h    = wsf + NN;                     // [N,64]
    float* g    = wsf + NN + (long)NN * OUTD;   // [N,64]

    float* o_seq = (float*)d_out;               // [T,N]
    float* z_seq = o_seq + (long)TT * NN;       // [T,N,64]

    k_init_deg   <<<(NN + 255) / 256, 256, 0, stream>>>(deg);
    k_accum_deg  <<<(EE + 255) / 256, 256, 0, stream>>>(ei, deg);
    k_deg_to_dinv<<<(NN + 255) / 256, 256, 0, stream>>>(deg);

    k_gemm_wmma  <<<NN / 16, 128, 0, stream>>>(x, W, h);   // N divisible by 16

    k_init_g     <<<((long)NN * OUTD + 255) / 256, 256, 0, stream>>>(deg, h, b, g);
    k_scatter    <<<((long)EE * OUTD + 255) / 256, 256, 0, stream>>>(ei, deg, h, g);

    k_lorentz    <<<(NN + 255) / 256, 256, 0, stream>>>(g, o_seq, z_seq);
}